// CTCBeamSearch_17789754541034
// MI455X (gfx1250) — compile-verified
//
#include <hip/hip_runtime.h>

#define NB       2          // batches
#define NBEAM    256
#define NV       128000
#define NVLM     127999     // V-1
#define NL       2
#define NH       2048
#define NROWS    512        // NB * NBEAM
#define KBEST    256
#define NBINS    4096
#define CAP      8192
#define NCHUNK   63         // ceil(NV / 2048): 62 full + one 1024 tail

// d_out float offsets
#define SC_OFF   0
#define HY_OFF   512
#define TK_OFF   1024
#define HG_OFF   1536
#define CG_OFF   (1536 + 2097152)

// ---------------- helpers ----------------

__device__ __forceinline__ unsigned key_of(float f) {
  unsigned u = __float_as_uint(f);
  return (u & 0x80000000u) ? ~u : (u | 0x80000000u);   // monotone float -> uint
}

__device__ __forceinline__ float key_to_float(unsigned k) {
  unsigned u = (k & 0x80000000u) ? (k ^ 0x80000000u) : ~k;
  return __uint_as_float(u);
}

// score of element v within row; rc0 = hyp - lse_ctc, lsl = lse_lm
// Identical op order everywhere -> bit-identical keys in hist & collect.
__device__ __forceinline__ float row_score(int v, const float* __restrict__ erow,
    const float* __restrict__ lrow, const float* __restrict__ prior,
    float rc0, float lsl) {
  float sc = erow[v] + rc0 - 0.3f * prior[v];
  if (v < NVLM) sc += 0.5f * (lrow[v] - lsl);   // padded lp_lm[V-1] == 0
  return sc;
}

__device__ __forceinline__ void lse_upd(float& m, float& s, float x) {
  if (x > m) { s = s * __expf(m - x) + 1.0f; m = x; }
  else       { s += __expf(x - m); }
}

// ---------------- pass 1: per-row logsumexp (deep async LDS pipeline) ----------------
// blocks 0..511: enc rows (len 128000) | blocks 512..1023: lm rows (len 127999)
__global__ __launch_bounds__(256) void lse_kernel(const float* __restrict__ enc,
                                                  const float* __restrict__ lm,
                                                  float* __restrict__ lse_out) {
  __shared__ __align__(16) float buf[2][4096];   // 2 x 16KB double buffer
  __shared__ float red_m[8], red_s[8];

  const int r   = blockIdx.x;
  const int tid = threadIdx.x;
  const bool is_lm = (r >= NROWS);
  const int row    = is_lm ? (r - NROWS) : r;
  const int rowlen = is_lm ? NVLM : NV;
  const float* base = is_lm ? (lm + (size_t)row * NVLM) : (enc + (size_t)row * NV);
  const int nfull = rowlen >> 12;                // 4096-float chunks (31 per row)

  float m = -3.402823466e+38f, s = 0.0f;

  const unsigned lb0 = (unsigned)(size_t)(&buf[0][0]);
  const unsigned lb1 = (unsigned)(size_t)(&buf[1][0]);
  const unsigned long long sbase = (unsigned long long)(size_t)base;

  // issue one 16KB chunk: 4 x b128 per lane -> 4 ASYNCcnt ticks per wave
  auto issue = [&](int c, int d) {
    unsigned gb = (unsigned)(c * 16384) + (unsigned)(tid * 16);
    unsigned lb = (d ? lb1 : lb0) + (unsigned)(tid * 16);
    #pragma unroll
    for (int k = 0; k < 4; ++k) {
      asm volatile("global_load_async_to_lds_b128 %0, %1, %2"
                   :: "v"(lb + (unsigned)(k * 4096)), "v"(gb + (unsigned)(k * 4096)),
                      "s"(sbase) : "memory");
    }
  };

  if (nfull > 0) issue(0, 0);
  for (int c = 0; c < nfull; ++c) {
    if (c + 1 < nfull) {
      issue(c + 1, (c + 1) & 1);
      asm volatile("s_wait_asynccnt %0" :: "i"(4));  // chunk c done, c+1 in flight
    } else {
      asm volatile("s_wait_asynccnt %0" :: "i"(0));
    }
    __syncthreads();
    const float* cb = buf[c & 1];
    #pragma unroll
    for (int k = 0; k < 4; ++k) {
      float4 x = *(const float4*)(cb + tid * 4 + k * 1024);
      lse_upd(m, s, x.x); lse_upd(m, s, x.y); lse_upd(m, s, x.z); lse_upd(m, s, x.w);
    }
    __syncthreads();                               // buffer reusable
  }
  for (int i = (nfull << 12) + tid; i < rowlen; i += 256)   // tail (<= 1024 elems)
    lse_upd(m, s, base[i]);

  #pragma unroll
  for (int o = 16; o > 0; o >>= 1) {
    float mo = __shfl_xor(m, o, 32);
    float so = __shfl_xor(s, o, 32);
    float mn = fmaxf(m, mo);
    s = s * __expf(m - mn) + so * __expf(mo - mn);
    m = mn;
  }
  if ((tid & 31) == 0) { red_m[tid >> 5] = m; red_s[tid >> 5] = s; }
  __syncthreads();
  if (tid == 0) {
    float M = red_m[0], S = red_s[0];
    for (int w = 1; w < 8; ++w) {
      float mo = red_m[w], so = red_s[w];
      float mn = fmaxf(M, mo);
      S = S * __expf(M - mn) + so * __expf(mo - mn);
      M = mn;
    }
    lse_out[r] = M + logf(S);
  }
}

// ---------------- pass 2: histogram + per-chunk max key ----------------
// grid (512 rows, 4 quarters); chunk = 2048 elements of v
__global__ __launch_bounds__(256) void hist_kernel(const float* __restrict__ enc,
    const float* __restrict__ lm, const float* __restrict__ prior,
    const float* __restrict__ hyp, const float* __restrict__ lse,
    unsigned* __restrict__ hist, unsigned* __restrict__ chunkmax) {
  __shared__ unsigned hloc[NBINS];
  __shared__ unsigned redmax[8];
  const int row = blockIdx.x;
  const int q   = blockIdx.y;
  const int b   = row >> 8;
  const int tid = threadIdx.x;
  for (int i = tid; i < NBINS; i += 256) hloc[i] = 0u;
  const float rc0 = hyp[row] - lse[row];
  const float lsl = lse[NROWS + row];
  const float* erow = enc + (size_t)row * NV;
  const float* lrow = lm  + (size_t)row * NVLM;
  const int c0 = q * 16, c1 = min(c0 + 16, NCHUNK);
  __syncthreads();
  for (int c = c0; c < c1; ++c) {
    const int vb = c << 11;
    unsigned kmax = 0u;
    #pragma unroll
    for (int k = 0; k < 8; ++k) {
      int v = vb + tid + (k << 8);
      if (v < NV) {
        unsigned key = key_of(row_score(v, erow, lrow, prior, rc0, lsl));
        atomicAdd(&hloc[key >> 20], 1u);
        kmax = max(kmax, key);
      }
    }
    #pragma unroll
    for (int o = 16; o > 0; o >>= 1)
      kmax = max(kmax, (unsigned)__shfl_xor((int)kmax, o, 32));
    if ((tid & 31) == 0) redmax[tid >> 5] = kmax;
    __syncthreads();
    if (tid == 0) {
      unsigned mx = redmax[0];
      #pragma unroll
      for (int w = 1; w < 8; ++w) mx = max(mx, redmax[w]);
      chunkmax[row * NCHUNK + c] = mx;
    }
    __syncthreads();
  }
  unsigned* hg = hist + (size_t)b * NBINS;
  for (int i = tid; i < NBINS; i += 256) {
    unsigned vv = hloc[i];
    if (vv) atomicAdd(&hg[i], vv);
  }
}

// ---------------- tiny: threshold bin per batch ----------------
__global__ void thresh_kernel(const unsigned* __restrict__ hist,
                              unsigned* __restrict__ thresh) {
  int b = threadIdx.x;
  if (b < NB) {
    const unsigned* hg = hist + (size_t)b * NBINS;
    int bin = NBINS;
    unsigned cum = 0;
    while (bin > 0 && cum < KBEST) { --bin; cum += hg[bin]; }
    thresh[b] = (unsigned)bin;
  }
}

// ---------------- pass 3 (pruned): rescan only qualifying chunks ----------------
__global__ __launch_bounds__(256) void collect_kernel(const float* __restrict__ enc,
    const float* __restrict__ lm, const float* __restrict__ prior,
    const float* __restrict__ hyp, const float* __restrict__ lse,
    const unsigned* __restrict__ thresh, const unsigned* __restrict__ chunkmax,
    unsigned* __restrict__ cnt, unsigned* __restrict__ ckey,
    unsigned* __restrict__ cidx) {
  const int row = blockIdx.x;
  const int b   = row >> 8;
  const int tid = threadIdx.x;
  const unsigned tb = thresh[b];
  const float rc0 = hyp[row] - lse[row];
  const float lsl = lse[NROWS + row];
  const float* erow = enc + (size_t)row * NV;
  const float* lrow = lm  + (size_t)row * NVLM;
  const unsigned beambase = (unsigned)((row & 255) * NV);
  for (int c = 0; c < NCHUNK; ++c) {
    if ((chunkmax[row * NCHUNK + c] >> 20) < tb) continue;   // chunk has no candidate
    const int vb = c << 11;
    #pragma unroll
    for (int k = 0; k < 8; ++k) {
      int v = vb + tid + (k << 8);
      if (v < NV) {
        unsigned key = key_of(row_score(v, erow, lrow, prior, rc0, lsl));
        if ((key >> 20) >= tb) {
          unsigned pos = atomicAdd(&cnt[b], 1u);
          if (pos < CAP) {
            ckey[(size_t)b * CAP + pos] = key;
            cidx[(size_t)b * CAP + pos] = beambase + (unsigned)v;
          }
        }
      }
    }
  }
}

// ---------------- sort candidates, emit scores / hypo / token ----------------
__global__ __launch_bounds__(1024) void sort_emit_kernel(const unsigned* __restrict__ cnt,
    const unsigned* __restrict__ ckey, const unsigned* __restrict__ cidx,
    float* __restrict__ out, int* __restrict__ hyp_ws) {
  __shared__ unsigned skey[CAP];
  __shared__ unsigned sidx[CAP];
  const int b = blockIdx.x;
  const unsigned n = min(cnt[b], (unsigned)CAP);
  for (int i = threadIdx.x; i < CAP; i += 1024) {
    if (i < (int)n) { skey[i] = ckey[(size_t)b * CAP + i]; sidx[i] = cidx[(size_t)b * CAP + i]; }
    else            { skey[i] = 0u;                        sidx[i] = 0xFFFFFFFFu; }
  }
  __syncthreads();
  // bitonic: descending by key, ascending index on ties (matches jax top_k)
  for (int k = 2; k <= CAP; k <<= 1) {
    for (int j = k >> 1; j > 0; j >>= 1) {
      for (int i = threadIdx.x; i < CAP; i += 1024) {
        int p = i ^ j;
        if (p > i) {
          unsigned long long a = ((unsigned long long)skey[i] << 32) | (unsigned)(~sidx[i]);
          unsigned long long c = ((unsigned long long)skey[p] << 32) | (unsigned)(~sidx[p]);
          bool dir = ((i & k) == 0);
          if ((a < c) == dir) {
            unsigned tk = skey[i]; skey[i] = skey[p]; skey[p] = tk;
            unsigned ti = sidx[i]; sidx[i] = sidx[p]; sidx[p] = ti;
          }
        }
      }
      __syncthreads();
    }
  }
  if (threadIdx.x < KBEST) {
    int t = threadIdx.x;
    unsigned fl = sidx[t];
    int hy  = (int)(fl / (unsigned)NV);
    int tok = (int)(fl - (unsigned)hy * (unsigned)NV);
    out[SC_OFF + (size_t)b * KBEST + t] = key_to_float(skey[t]);
    out[HY_OFF + (size_t)b * KBEST + t] = (float)hy;    // exact (<= 255)
    out[TK_OFF + (size_t)b * KBEST + t] = (float)tok;   // exact (< 2^24)
    hyp_ws[b * KBEST + t] = hy;
  }
}

// ---------------- gather lm_h / lm_c rows ----------------
__global__ __launch_bounds__(256) void gather_kernel(const float* __restrict__ lm_h,
    const float* __restrict__ lm_c, const int* __restrict__ hyp_ws,
    float* __restrict__ out) {
  const int k  = blockIdx.x;        // nbest slot
  const int bl = blockIdx.y;        // b*L + l
  const int b  = bl >> 1;
  const int hy = hyp_ws[b * KBEST + k];
  const float* srcH = lm_h + (((size_t)bl * NBEAM + hy) * NH);
  const float* srcC = lm_c + (((size_t)bl * NBEAM + hy) * NH);
  float* dstH = out + HG_OFF + (((size_t)bl * KBEST + k) * NH);
  float* dstC = out + CG_OFF + (((size_t)bl * KBEST + k) * NH);
  for (int i = threadIdx.x * 4; i < NH; i += 256 * 4) {
    *(float4*)(dstH + i) = *(const float4*)(srcH + i);
    *(float4*)(dstC + i) = *(const float4*)(srcC + i);
  }
}

// ---------------- host launch ----------------
extern "C" void kernel_launch(void* const* d_in, const int* in_sizes, int n_in,
                              void* d_out, int out_size, void* d_ws, size_t ws_size,
                              hipStream_t stream) {
  (void)in_sizes; (void)n_in; (void)out_size; (void)ws_size;
  const float* enc   = (const float*)d_in[0];
  const float* lmo   = (const float*)d_in[1];
  const float* prior = (const float*)d_in[2];
  const float* hyp   = (const float*)d_in[3];
  const float* lmh   = (const float*)d_in[4];
  const float* lmc   = (const float*)d_in[5];
  float* out = (float*)d_out;

  // ws layout (u32 units):
  // [0,1024) lse | [1024,9216) hist | [9216,9218) thresh | [9218,9220) cnt
  // [9280,25664) cand keys | [25664,42048) cand idx | [42048,42560) hypo int
  // [42560,74816) chunk max keys (512 rows x 63 chunks)
  float*    lse    = (float*)d_ws;
  unsigned* hist   = (unsigned*)d_ws + 1024;
  unsigned* thresh = (unsigned*)d_ws + 9216;
  unsigned* cnt    = (unsigned*)d_ws + 9218;
  unsigned* ckey   = (unsigned*)d_ws + 9280;
  unsigned* cidx   = (unsigned*)d_ws + 25664;
  int*      hypws  = (int*)d_ws + 42048;
  unsigned* cmax   = (unsigned*)d_ws + 42560;

  hipMemsetAsync((char*)d_ws + 1024 * 4, 0, (9220 - 1024) * 4, stream);

  lse_kernel    <<<1024, 256, 0, stream>>>(enc, lmo, lse);
  hist_kernel   <<<dim3(NROWS, 4), 256, 0, stream>>>(enc, lmo, prior, hyp, lse, hist, cmax);
  thresh_kernel <<<1, 32, 0, stream>>>(hist, thresh);
  collect_kernel<<<NROWS, 256, 0, stream>>>(enc, lmo, prior, hyp, lse,
                                            thresh, cmax, cnt, ckey, cidx);
  sort_emit_kernel<<<NB, 1024, 0, stream>>>(cnt, ckey, cidx, out, hypws);
  gather_kernel <<<dim3(KBEST, NB * NL), 256, 0, stream>>>(lmh, lmc, hypws, out);
}